// XAS_Mask_Structure_41841571397769
// MI455X (gfx1250) — compile-verified
//
#include <hip/hip_runtime.h>

// Problem constants (match reference).
constexpr int Nn = 10000;   // nodes
constexpr int Ne = 320000;  // edges
constexpr int Ng = 64;      // graphs
constexpr int D  = 256;     // embedding dim
constexpr int Ad = 118;     // atomic_num dim
constexpr int Od = 100;     // output dim

typedef __attribute__((ext_vector_type(2))) float v2f;
typedef __attribute__((ext_vector_type(8))) float v8f;

// D = A(16x4, f32) * B(4x16, f32) + C(16x16, f32)
__device__ __forceinline__ v8f wmma_k4(v2f a, v2f b, v8f c) {
  return __builtin_amdgcn_wmma_f32_16x16x4_f32(false, a, false, b, (short)0, c,
                                               false, false);
}

// ---------------------------------------------------------------- utilities
__global__ void zero_f32(float* __restrict__ p, int n) {
  int i = blockIdx.x * blockDim.x + threadIdx.x;
  if (i < n) p[i] = 0.0f;
}

// --------------------------------------------- h = atomic_num @ W_node + b
// Register-blocked 1x4: one wave computes a 16x64 output strip; the A
// fragment is loaded once per k-step and feeds 4 WMMAs. 625 M-tiles x
// 4 N-groups = 2500 wave-jobs = 625 blocks x 4 waves (EXEC always all-ones).
// K=118: 29 unguarded k-steps (K=0..115) + one branchless clamped tail —
// no divergent guards around loads.
__global__ void node_embed_wmma(const float* __restrict__ A,
                                const float* __restrict__ W,
                                const float* __restrict__ bias,
                                float* __restrict__ H) {
  const int wave = threadIdx.x >> 5;
  const int lane = threadIdx.x & 31;
  const int job  = blockIdx.x * 4 + wave;     // 0..2499
  const int mt   = job >> 2;                  // 0..624
  const int ntg  = job & 3;                   // 0..3 (cols ntg*64 .. +63)
  const int half = lane >> 4;                 // 0: K lo pair, 1: K hi pair
  const int l15  = lane & 15;
  const int arow = mt * 16 + l15;             // < 10000 always
  const int col0 = ntg * 64 + l15;
  const float* Arow = A + (size_t)arow * Ad;
  v8f acc0 = {}, acc1 = {}, acc2 = {}, acc3 = {};
  for (int kb = 0; kb < 29; ++kb) {           // K = 0..115, all in-bounds
    const int k0 = kb * 4 + half * 2;
    v2f a, b;
    a.x = Arow[k0];
    a.y = Arow[k0 + 1];
    const float* Wk0 = W + (size_t)k0 * D;
    b.x = Wk0[col0];       b.y = Wk0[D + col0];
    acc0 = wmma_k4(a, b, acc0);
    b.x = Wk0[col0 + 16];  b.y = Wk0[D + col0 + 16];
    acc1 = wmma_k4(a, b, acc1);
    b.x = Wk0[col0 + 32];  b.y = Wk0[D + col0 + 32];
    acc2 = wmma_k4(a, b, acc2);
    b.x = Wk0[col0 + 48];  b.y = Wk0[D + col0 + 48];
    acc3 = wmma_k4(a, b, acc3);
  }
  {                                           // tail: K=116,117 (half=0 only)
    const float sel = half ? 0.0f : 1.0f;     // half=1 lanes contribute zero
    v2f a, b;
    a.x = Arow[116] * sel;                    // clamped, always in-bounds
    a.y = Arow[117] * sel;
    const float* Wk0 = W + (size_t)116 * D;
    b.x = Wk0[col0];       b.y = Wk0[D + col0];
    acc0 = wmma_k4(a, b, acc0);
    b.x = Wk0[col0 + 16];  b.y = Wk0[D + col0 + 16];
    acc1 = wmma_k4(a, b, acc1);
    b.x = Wk0[col0 + 32];  b.y = Wk0[D + col0 + 32];
    acc2 = wmma_k4(a, b, acc2);
    b.x = Wk0[col0 + 48];  b.y = Wk0[D + col0 + 48];
    acc3 = wmma_k4(a, b, acc3);
  }
#pragma unroll
  for (int v = 0; v < 8; ++v) {
    const int row = mt * 16 + v + half * 8;
    float* Hrow = H + (size_t)row * D;
    Hrow[col0]      = acc0[v] + bias[col0];
    Hrow[col0 + 16] = acc1[v] + bias[col0 + 16];
    Hrow[col0 + 32] = acc2[v] + bias[col0 + 32];
    Hrow[col0 + 48] = acc3[v] + bias[col0 + 48];
  }
}

// ------------------- m = relu(h[src] + len*W_edge + b_edge); atomic -> agg
// e is a rank-1 outer product: recompute per element instead of
// materializing 82M floats. 64 threads/edge, float4 per thread. Prefetch the
// gather row of an edge ~32 ahead to hide L2 latency behind the atomics.
__global__ void gine_message(const float* __restrict__ H,
                             const float* __restrict__ length,
                             const int* __restrict__ src,
                             const int* __restrict__ dst,
                             const float* __restrict__ We,
                             const float* __restrict__ be,
                             float* __restrict__ agg) {
  const int idx = blockIdx.x * blockDim.x + threadIdx.x;
  const int e  = idx >> 6;          // D/4 = 64 threads per edge
  const int d4 = (idx & 63) << 2;
  if (e >= Ne) return;
  if (e + 32 < Ne) {                // emits global_prefetch_b8
    __builtin_prefetch(H + (size_t)src[e + 32] * D + d4, 0, 3);
  }
  const int s = src[e];
  const int t = dst[e];
  const float len = length[e];
  const float4 hv = *(const float4*)(H  + (size_t)s * D + d4);
  const float4 wv = *(const float4*)(We + d4);
  const float4 bv = *(const float4*)(be + d4);
  const float m0 = fmaxf(hv.x + len * wv.x + bv.x, 0.0f);
  const float m1 = fmaxf(hv.y + len * wv.y + bv.y, 0.0f);
  const float m2 = fmaxf(hv.z + len * wv.z + bv.z, 0.0f);
  const float m3 = fmaxf(hv.w + len * wv.w + bv.w, 0.0f);
  float* base = agg + (size_t)t * D + d4;
  atomicAdd(base + 0, m0);
  atomicAdd(base + 1, m1);
  atomicAdd(base + 2, m2);
  atomicAdd(base + 3, m3);
}

// -------------------- h' = ((1+eps)*h + agg) @ W_gnn + b_gnn  (fused A-load)
// 1x4 register blocking: A fragment ((1+eps)h+agg) loaded once, reused by
// 4 WMMAs. K=256, no tail, no guards anywhere.
__global__ void gine_update_wmma(const float* __restrict__ Hin,
                                 const float* __restrict__ Agg,
                                 const float* __restrict__ Wg,
                                 const float* __restrict__ bg,
                                 const float* __restrict__ eps_ptr,
                                 float* __restrict__ Hout) {
  const int wave = threadIdx.x >> 5;
  const int lane = threadIdx.x & 31;
  const int job  = blockIdx.x * 4 + wave;     // 0..2499
  const int mt   = job >> 2;
  const int ntg  = job & 3;
  const int half = lane >> 4;
  const int l15  = lane & 15;
  const int arow = mt * 16 + l15;
  const int col0 = ntg * 64 + l15;
  const float sc = 1.0f + eps_ptr[0];
  const float* Hrow = Hin + (size_t)arow * D;
  const float* Grow = Agg + (size_t)arow * D;
  v8f acc0 = {}, acc1 = {}, acc2 = {}, acc3 = {};
  for (int kb = 0; kb < D / 4; ++kb) {        // K = 256
    const int k0 = kb * 4 + half * 2;
    v2f a, b;
    a.x = sc * Hrow[k0]     + Grow[k0];
    a.y = sc * Hrow[k0 + 1] + Grow[k0 + 1];
    const float* Wk0 = Wg + (size_t)k0 * D;
    b.x = Wk0[col0];       b.y = Wk0[D + col0];
    acc0 = wmma_k4(a, b, acc0);
    b.x = Wk0[col0 + 16];  b.y = Wk0[D + col0 + 16];
    acc1 = wmma_k4(a, b, acc1);
    b.x = Wk0[col0 + 32];  b.y = Wk0[D + col0 + 32];
    acc2 = wmma_k4(a, b, acc2);
    b.x = Wk0[col0 + 48];  b.y = Wk0[D + col0 + 48];
    acc3 = wmma_k4(a, b, acc3);
  }
#pragma unroll
  for (int v = 0; v < 8; ++v) {
    const int row = mt * 16 + v + half * 8;
    float* Orow = Hout + (size_t)row * D;
    Orow[col0]      = acc0[v] + bg[col0];
    Orow[col0 + 16] = acc1[v] + bg[col0 + 16];
    Orow[col0 + 32] = acc2[v] + bg[col0 + 32];
    Orow[col0 + 48] = acc3[v] + bg[col0 + 48];
  }
}

// ----------------------------------------- feat[g] = sum over nodes of h[n]
__global__ void pool_kernel(const float* __restrict__ H,
                            const int* __restrict__ gid,
                            float* __restrict__ feat) {
  const int idx = blockIdx.x * blockDim.x + threadIdx.x;
  const int n  = idx >> 6;
  const int d4 = (idx & 63) << 2;
  if (n >= Nn) return;
  const int g = gid[n];
  const float4 hv = *(const float4*)(H + (size_t)n * D + d4);
  float* base = feat + (size_t)g * D + d4;
  atomicAdd(base + 0, hv.x);
  atomicAdd(base + 1, hv.y);
  atomicAdd(base + 2, hv.z);
  atomicAdd(base + 3, hv.w);
}

// ------------------------------------ global min/max over feat (16K floats)
__global__ void minmax_kernel(const float* __restrict__ feat,
                              float* __restrict__ red) {
  __shared__ float smn[256];
  __shared__ float smx[256];
  float mn = 3.402823466e+38f, mx = -3.402823466e+38f;
  for (int i = threadIdx.x; i < Ng * D; i += 256) {
    const float v = feat[i];
    mn = fminf(mn, v);
    mx = fmaxf(mx, v);
  }
  smn[threadIdx.x] = mn;
  smx[threadIdx.x] = mx;
  __syncthreads();
  for (int s = 128; s > 0; s >>= 1) {
    if (threadIdx.x < (unsigned)s) {
      smn[threadIdx.x] = fminf(smn[threadIdx.x], smn[threadIdx.x + s]);
      smx[threadIdx.x] = fmaxf(smx[threadIdx.x], smx[threadIdx.x + s]);
    }
    __syncthreads();
  }
  if (threadIdx.x == 0) { red[0] = smn[0]; red[1] = smx[0]; }
}

// ------- out = ((feat - mn) * inv) @ W_mlp + b_mlp  (normalization fused)
// M=64 -> 4 row tiles, N=100 -> 7 col tiles (pad to 112). 28 tiles,
// 4 waves/block -> 7 blocks exactly. Out-of-range B columns use a clamped
// always-in-bounds address times a 0/1 selector (branchless); only the
// final store is predicated.
__global__ void head_wmma(const float* __restrict__ feat,
                          const float* __restrict__ Wm,
                          const float* __restrict__ bm,
                          const float* __restrict__ red,
                          const float* __restrict__ eps_param,
                          float* __restrict__ out) {
  const int wave = threadIdx.x >> 5;
  const int lane = threadIdx.x & 31;
  const int tile = blockIdx.x * 4 + wave;
  const int NT = 7;
  const int mt = tile / NT;
  const int nt = tile % NT;
  const int half = lane >> 4;
  const int l15  = lane & 15;
  const int arow = mt * 16 + l15;              // < 64 always
  const int bcol = nt * 16 + l15;              // may exceed 99
  const int bcc  = bcol < Od ? bcol : (Od - 1);  // clamped address
  const float sel = bcol < Od ? 1.0f : 0.0f;     // branchless column mask
  const float mn  = red[0];
  const float inv = 1.0f / (eps_param[0] + red[1] - mn);
  const float* Frow = feat + (size_t)arow * D;
  v8f c = {};
  for (int kb = 0; kb < D / 4; ++kb) {
    const int k0 = kb * 4 + half * 2;
    v2f a, b;
    a.x = (Frow[k0]     - mn) * inv;
    a.y = (Frow[k0 + 1] - mn) * inv;
    b.x = Wm[(size_t)(k0)     * Od + bcc] * sel;
    b.y = Wm[(size_t)(k0 + 1) * Od + bcc] * sel;
    c = wmma_k4(a, b, c);
  }
#pragma unroll
  for (int v = 0; v < 8; ++v) {
    const int row = mt * 16 + v + half * 8;
    if (bcol < Od) out[(size_t)row * Od + bcol] = c[v] + bm[bcol];
  }
}

extern "C" void kernel_launch(void* const* d_in, const int* in_sizes, int n_in,
                              void* d_out, int out_size, void* d_ws,
                              size_t ws_size, hipStream_t stream) {
  const float* atomic_num = (const float*)d_in[0];
  const float* length     = (const float*)d_in[1];
  const int*   src        = (const int*)d_in[2];
  const int*   dst        = (const int*)d_in[3];
  const int*   graph_ids  = (const int*)d_in[4];
  const float* W_node     = (const float*)d_in[5];
  const float* b_node     = (const float*)d_in[6];
  const float* W_edge     = (const float*)d_in[7];
  const float* b_edge     = (const float*)d_in[8];
  const float* gine_eps   = (const float*)d_in[9];
  const float* W_gnn      = (const float*)d_in[10];
  const float* b_gnn      = (const float*)d_in[11];
  const float* eps_param  = (const float*)d_in[12];
  const float* W_mlp      = (const float*)d_in[13];
  const float* b_mlp      = (const float*)d_in[14];
  float* out = (float*)d_out;

  // Workspace layout (~31 MB; entire set fits in MI455X's 192 MB L2).
  const size_t ND = (size_t)Nn * D;
  float* h    = (float*)d_ws;
  float* agg  = h + ND;
  float* h2   = agg + ND;
  float* feat = h2 + ND;
  float* red  = feat + (size_t)Ng * D;

  // 1. node embedding GEMM: 2500 wave-jobs / 4 waves = 625 blocks.
  node_embed_wmma<<<625, 128, 0, stream>>>(atomic_num, W_node, b_node, h);

  // 2. two GINEConv layers.
  float* cur = h;
  float* nxt = h2;
  const int msg_blocks = (Ne * 64 + 255) / 256;   // 80000
  for (int layer = 0; layer < 2; ++layer) {
    zero_f32<<<(int)((ND + 255) / 256), 256, 0, stream>>>(agg, (int)ND);
    gine_message<<<msg_blocks, 256, 0, stream>>>(cur, length, src, dst,
                                                 W_edge, b_edge, agg);
    gine_update_wmma<<<625, 128, 0, stream>>>(
        cur, agg, W_gnn + (size_t)layer * D * D, b_gnn + (size_t)layer * D,
        gine_eps + layer, nxt);
    float* t = cur; cur = nxt; nxt = t;
  }

  // 3. per-graph pooling.
  zero_f32<<<(Ng * D + 255) / 256, 256, 0, stream>>>(feat, Ng * D);
  pool_kernel<<<(Nn * 64 + 255) / 256, 256, 0, stream>>>(cur, graph_ids, feat);

  // 4. global min/max.
  minmax_kernel<<<1, 256, 0, stream>>>(feat, red);

  // 5. normalized head GEMM: 28 tiles / 4 waves = 7 blocks.
  head_wmma<<<7, 128, 0, stream>>>(feat, W_mlp, b_mlp, red, eps_param, out);
}